// RNN_30365418782869
// MI455X (gfx1250) — compile-verified
//
#include <hip/hip_runtime.h>
#include <math.h>

// Problem constants (match reference)
#define S_LEN 128
#define B_SZ  128
#define V_SZ  10000
#define H_SZ  512
#define E_SZ  512
#define K_REC 1024   // H + E
#define L_LAY 2

typedef float v2f __attribute__((ext_vector_type(2)));
typedef float v4f __attribute__((ext_vector_type(4)));
typedef float v8f __attribute__((ext_vector_type(8)));

__device__ __forceinline__ v2f ld2(const float* __restrict__ p) {
    return *(const v2f*)p;   // 8B aligned by construction (even k offsets)
}

// ---------------------------------------------------------------------------
// Recurrent layer: hout = tanh( [hprev | x] @ W^T + bias )
//   MODE 0: x = emb[tok] * scale   (layer 0)
//   MODE 1: x = xin                (layer >= 1)
// C is [B=128, H=512] -> 8 x 32 tiles of 16x16, one wave per tile.
// ---------------------------------------------------------------------------
template<int MODE, bool WTOPS>
__global__ void rnn_layer_kernel(
    const float* __restrict__ hprev,   // [B, H]
    const float* __restrict__ xin,     // [B, E]   (MODE 1)
    const int*   __restrict__ tok,     // [B]      (MODE 0)
    const float* __restrict__ emb,     // [V, E]   (MODE 0)
    float scale,
    const float* __restrict__ W,       // [H, H+E] row-major
    const float* __restrict__ bias,    // [H]
    float* __restrict__ hout,          // [B, H]
    float* __restrict__ tops)          // [B, H] second copy (WTOPS)
{
    const int lane = threadIdx.x & 31;
    const int wave = (blockIdx.x * (blockDim.x >> 5)) + (threadIdx.x >> 5);
    const int tileM = wave & 7;        // 8 tiles over B=128
    const int tileN = wave >> 3;       // 32 tiles over H=512
    const int l16   = lane & 15;
    const int half  = lane >> 4;       // 0: K pair {0,1}, 1: K pair {2,3}

    const int row = tileM * 16 + l16;
    const int col = tileN * 16 + l16;

    const float* __restrict__ ah = hprev + (size_t)row * H_SZ + 2 * half;
    const float* __restrict__ ax;
    if (MODE == 0) {
        const int t = tok[row];
        ax = emb + (size_t)t * E_SZ + 2 * half;
    } else {
        ax = xin + (size_t)row * E_SZ + 2 * half;
    }
    const float* __restrict__ wh = W + (size_t)col * K_REC + 2 * half;
    const float* __restrict__ wx = wh + H_SZ;

    v8f acc = {0.f, 0.f, 0.f, 0.f, 0.f, 0.f, 0.f, 0.f};

#pragma unroll 8
    for (int kk = 0; kk < H_SZ; kk += 4) {
        v2f a = ld2(ah + kk);
        v2f b = ld2(wh + kk);
        acc = __builtin_amdgcn_wmma_f32_16x16x4_f32(
            false, a, false, b, (short)0, acc, false, false);
    }
#pragma unroll 8
    for (int kk = 0; kk < E_SZ; kk += 4) {
        v2f a = ld2(ax + kk);
        if (MODE == 0) { a.x *= scale; a.y *= scale; }
        v2f b = ld2(wx + kk);
        acc = __builtin_amdgcn_wmma_f32_16x16x4_f32(
            false, a, false, b, (short)0, acc, false, false);
    }

    const float bv = bias[col];
#pragma unroll
    for (int r = 0; r < 8; ++r) {
        const int rr = tileM * 16 + r + half * 8;   // C/D row layout
        const float v = tanhf(acc[r] + bv);
        hout[(size_t)rr * H_SZ + col] = v;
        if (WTOPS) tops[(size_t)rr * H_SZ + col] = v;
    }
}

// ---------------------------------------------------------------------------
// Output projection: logits[S*B, V] = tops[S*B, H] @ Wout^T + bout
// LDS-staged: block (8 waves) computes a 128x80 C tile.
//   M: 16384 = 128 blocks * 128 rows, N: 10000 = 125 blocks * 80 cols (exact).
//   K chunks of 64, A/B panels staged in LDS with row stride 68 floats
//   (16B-aligned rows for b128 stores; conflict-free b64 fragment reads).
//   Next chunk's global loads are issued into registers before computing the
//   current chunk from LDS (LOADcnt hides global latency behind WMMAs).
// Wave w handles rows [w*16, w*16+16) x all 80 cols: 5 accumulators.
// ---------------------------------------------------------------------------
#define KC     64
#define LDA    68                 // padded row stride (floats)
#define NBLK   80
#define MBLK   128

__global__ __launch_bounds__(256) void out_proj_kernel(
    const float* __restrict__ tops,    // [S*B, H]
    const float* __restrict__ Wout,    // [V, H] row-major
    const float* __restrict__ bout,    // [V]
    float* __restrict__ logits)        // [S*B, V]
{
    __shared__ float sA[MBLK * LDA];   // 128*68*4 = 34816 B
    __shared__ float sB[NBLK * LDA];   //  80*68*4 = 21760 B

    const int tid  = threadIdx.x;
    const int lane = tid & 31;
    const int wv   = tid >> 5;         // wave 0..7
    const int l16  = lane & 15;
    const int half = lane >> 4;

    const int nb = blockIdx.x;         // 0..124
    const int mb = blockIdx.y;         // 0..127

    const float* __restrict__ aG = tops + (size_t)mb * MBLK * H_SZ;
    const float* __restrict__ bG = Wout + (size_t)nb * NBLK * H_SZ;

    // Staging registers: A 8 x float4/thread, B 5 x float4/thread.
    v4f ra[8], rb[5];

    // ---- helpers as lambdas (fully inlined) ----
    auto load_regs = [&](int kc) {
#pragma unroll
        for (int i = 0; i < 8; ++i) {
            const int f = tid + i * 256;           // 0..2047
            const int r = f >> 4, c = (f & 15) * 4;
            ra[i] = *(const v4f*)(aG + (size_t)r * H_SZ + kc + c);
        }
#pragma unroll
        for (int i = 0; i < 5; ++i) {
            const int f = tid + i * 256;           // 0..1279
            const int r = f >> 4, c = (f & 15) * 4;
            rb[i] = *(const v4f*)(bG + (size_t)r * H_SZ + kc + c);
        }
    };
    auto store_smem = [&]() {
#pragma unroll
        for (int i = 0; i < 8; ++i) {
            const int f = tid + i * 256;
            const int r = f >> 4, c = (f & 15) * 4;
            *(v4f*)(&sA[r * LDA + c]) = ra[i];     // 16B aligned (68*4=272=16*17)
        }
#pragma unroll
        for (int i = 0; i < 5; ++i) {
            const int f = tid + i * 256;
            const int r = f >> 4, c = (f & 15) * 4;
            *(v4f*)(&sB[r * LDA + c]) = rb[i];
        }
    };

    v8f acc[5];
#pragma unroll
    for (int n = 0; n < 5; ++n)
        acc[n] = (v8f){0.f,0.f,0.f,0.f,0.f,0.f,0.f,0.f};

    float bv[5];
#pragma unroll
    for (int n = 0; n < 5; ++n)
        bv[n] = bout[nb * NBLK + n * 16 + l16];

    const float* pa  = &sA[(wv * 16 + l16) * LDA + 2 * half];
    const float* pb0 = &sB[l16 * LDA + 2 * half];

    load_regs(0);
    store_smem();
    __syncthreads();

    for (int c = 0; c < H_SZ / KC; ++c) {          // 8 chunks
        if (c + 1 < H_SZ / KC) load_regs((c + 1) * KC);  // global loads in flight

#pragma unroll 4
        for (int kk = 0; kk < KC; kk += 4) {
            v2f a = *(const v2f*)(pa + kk);
#pragma unroll
            for (int n = 0; n < 5; ++n) {
                v2f b = *(const v2f*)(pb0 + n * 16 * LDA + kk);
                acc[n] = __builtin_amdgcn_wmma_f32_16x16x4_f32(
                    false, a, false, b, (short)0, acc[n], false, false);
            }
        }

        if (c + 1 < H_SZ / KC) {
            __syncthreads();                        // everyone done reading LDS
            store_smem();                           // next chunk -> LDS
            __syncthreads();
        }
    }

    // Epilogue: bias add + store 128x80 tile.
#pragma unroll
    for (int n = 0; n < 5; ++n) {
        const int col = nb * NBLK + n * 16 + l16;
#pragma unroll
        for (int r = 0; r < 8; ++r) {
            const int row = mb * MBLK + wv * 16 + r + half * 8;
            logits[(long long)row * V_SZ + col] = acc[n][r] + bv[n];
        }
    }
}

// ---------------------------------------------------------------------------
extern "C" void kernel_launch(void* const* d_in, const int* in_sizes, int n_in,
                              void* d_out, int out_size, void* d_ws, size_t ws_size,
                              hipStream_t stream) {
    const int*   inputs = (const int*)  d_in[0];   // [S, B]
    const float* hidden = (const float*)d_in[1];   // [L, B, H]
    const float* emb    = (const float*)d_in[2];   // [V, E]
    const float* W0     = (const float*)d_in[3];   // [H, H+E]
    const float* b0     = (const float*)d_in[4];   // [H]
    const float* Wl     = (const float*)d_in[5];   // [L-1, H, 2H]
    const float* bl     = (const float*)d_in[6];   // [L-1, H]
    const float* Wout   = (const float*)d_in[7];   // [V, H]
    const float* bout   = (const float*)d_in[8];   // [V]
    float* out = (float*)d_out;                    // logits [S,B,V] ++ h_final [L,B,H]

    const float scale = sqrtf((float)E_SZ);
    const size_t LBH = (size_t)L_LAY * B_SZ * H_SZ;
    const size_t BH  = (size_t)B_SZ * H_SZ;

    float* wsf    = (float*)d_ws;
    float* hst[2] = { wsf, wsf + LBH };            // parity ping-pong
    float* tops   = wsf + 2 * LBH;                 // [S, B, H]

    hipMemcpyAsync(hst[0], hidden, LBH * sizeof(float),
                   hipMemcpyDeviceToDevice, stream);

    dim3 blk(256);
    dim3 grd_rec(32);                              // 256 waves = 8x32 tiles

    for (int t = 0; t < S_LEN; ++t) {
        const int p = t & 1, pn = p ^ 1;
        rnn_layer_kernel<0, false><<<grd_rec, blk, 0, stream>>>(
            hst[p] + 0, nullptr, inputs + t * B_SZ, emb, scale,
            W0, b0, hst[pn] + 0, nullptr);
        rnn_layer_kernel<1, true><<<grd_rec, blk, 0, stream>>>(
            hst[p] + BH, hst[pn] + 0, nullptr, nullptr, 1.0f,
            Wl, bl, hst[pn] + BH, tops + (size_t)t * BH);
    }

    dim3 grd_out(V_SZ / NBLK, (S_LEN * B_SZ) / MBLK);   // 125 x 128 blocks
    out_proj_kernel<<<grd_out, blk, 0, stream>>>(tops, Wout, bout, out);

    hipMemcpyAsync(out + (size_t)S_LEN * B_SZ * V_SZ, hst[0],
                   LBH * sizeof(float), hipMemcpyDeviceToDevice, stream);
}